// ClusteringLayer_65146063946398
// MI455X (gfx1250) — compile-verified
//
#include <hip/hip_runtime.h>

typedef __attribute__((ext_vector_type(2))) float v2f;
typedef __attribute__((ext_vector_type(8))) float v8f;

#define T_ 16
#define V_ 1024
#define D_ 128
#define B_ 2048
#define CAP_ 2304       // 2048 + 16 groups * up to 15 pad, rounded to /16
#define MAXTILES_ (CAP_/16)

// ---- d_ws byte offsets (all >=16B aligned where vector loads occur) ----
#define WS_COUNTS   0        // int[16]
#define WS_CURSOR   64       // int[16]
#define WS_BASE     128      // int[17] (padded exclusive prefix, units = rows)
#define WS_PERM     256      // int[CAP_]
#define WS_BESTIDX  9472     // int[B_]
#define WS_RNORM    17664    // float[T_*V_]
#define WS_XG       83200    // float[CAP_*D_]  (83200 % 16 == 0)

__global__ void k_init(int* perm, int* counts, int* cursor, float* xg) {
  int i = blockIdx.x * blockDim.x + threadIdx.x;
  if (i < CAP_ * D_) xg[i] = 0.0f;
  if (i < CAP_) perm[i] = -1;
  if (i < T_) { counts[i] = 0; cursor[i] = 0; }
}

__global__ void k_count(const int* __restrict__ idx, int* counts) {
  int b = blockIdx.x * blockDim.x + threadIdx.x;
  if (b < B_) atomicAdd(&counts[idx[b] & (T_ - 1)], 1);
}

__global__ void k_base(const int* __restrict__ counts, int* base) {
  int acc = 0;
  for (int t = 0; t < T_; ++t) { base[t] = acc; acc += (counts[t] + 15) & ~15; }
  base[T_] = acc;
}

__global__ void k_scatter(const int* __restrict__ idx, const int* __restrict__ base,
                          int* cursor, int* perm) {
  int b = blockIdx.x * blockDim.x + threadIdx.x;
  if (b < B_) {
    int t = idx[b] & (T_ - 1);
    int slot = base[t] + atomicAdd(&cursor[t], 1);
    perm[slot] = b;
  }
}

__global__ void k_gatherx(const int* __restrict__ perm, const float* __restrict__ inputs,
                          float* __restrict__ xg) {
  int slot = blockIdx.x;
  int d = threadIdx.x;
  int b = perm[slot];
  if (b >= 0) xg[slot * D_ + d] = inputs[b * D_ + d];
}

// one wave (32 lanes) per (t,v) row: lane loads float4, wave-reduce sum of squares
__global__ void k_rnorm(const float* __restrict__ latent, float* __restrict__ rnorm) {
  int wave = threadIdx.x >> 5;
  int lane = threadIdx.x & 31;
  int row = blockIdx.x * 8 + wave;                  // grid 2048 * 8 waves = 16384 rows
  const float4* p = (const float4*)(latent + (size_t)row * D_);
  float4 q = p[lane];
  float ss = q.x * q.x + q.y * q.y + q.z * q.z + q.w * q.w;
  #pragma unroll
  for (int m = 16; m >= 1; m >>= 1) ss += __shfl_xor(ss, m, 32);
  if (lane == 0) rnorm[row] = rsqrtf(fmaxf(ss, 1e-12f));
}

// one block = one 16-batch tile of a group; 8 waves split 64 v-tiles; K=128 via 32 wmma_f32_16x16x4
__global__ __launch_bounds__(256) void k_sim_argmax(
    const float* __restrict__ latent, const float* __restrict__ xg,
    const float* __restrict__ rnorm, const int* __restrict__ base,
    const int* __restrict__ perm, int* __restrict__ bestidx) {
  const int tile = blockIdx.x;
  const int totalTiles = base[T_] >> 4;
  if (tile >= totalTiles) return;                   // block-uniform exit (EXEC stays full)

  const int row0 = tile * 16;
  int t = 0;
  #pragma unroll
  for (int i = 0; i < T_; ++i) if (base[i] <= row0) t = i;   // owning group

  const int wave = threadIdx.x >> 5;
  const int lane = threadIdx.x & 31;
  const int hi   = lane >> 4;                       // K-half selector
  const int l    = lane & 15;                       // M (v-row) for A, N (batch col) for B

  const float* __restrict__ xrow  = xg + (size_t)(row0 + l) * D_ + 2 * hi;
  const float* __restrict__ lat_t = latent + (size_t)t * V_ * D_;
  const float* __restrict__ rn_t  = rnorm + t * V_;

  float bestval = -3.402823466e38f;
  int   bestv   = 0x40000000;

  for (int vt = wave; vt < V_ / 16; vt += 8) {      // wave-uniform loop
    const float* arow = lat_t + (size_t)(vt * 16 + l) * D_ + 2 * hi;
    v8f c = {};
    #pragma unroll
    for (int k = 0; k < D_; k += 4) {
      v2f a = *(const v2f*)(arow + k);              // A: 16 latent rows x K=4
      v2f b = *(const v2f*)(xrow + k);              // B: K=4 x 16 batch cols
      c = __builtin_amdgcn_wmma_f32_16x16x4_f32(false, a, false, b,
                                                (short)0, c, false, false);
    }
    #pragma unroll
    for (int r = 0; r < 8; ++r) {                   // lane holds 8 M-rows for batch N=l
      int v = vt * 16 + hi * 8 + r;
      float val = c[r] * rn_t[v];
      if (val > bestval || (val == bestval && v < bestv)) { bestval = val; bestv = v; }
    }
  }
  // merge the two lanes (l, l+16) that share batch column l
  float oval = __shfl_xor(bestval, 16, 32);
  int   ov   = __shfl_xor(bestv, 16, 32);
  if (oval > bestval || (oval == bestval && ov < bestv)) { bestval = oval; bestv = ov; }

  __shared__ float sval[8][16];
  __shared__ int   sv[8][16];
  if (hi == 0) { sval[wave][l] = bestval; sv[wave][l] = bestv; }
  __syncthreads();
  if (threadIdx.x < 16) {
    float bv = sval[0][threadIdx.x];
    int   bi = sv[0][threadIdx.x];
    #pragma unroll
    for (int w = 1; w < 8; ++w) {
      float v2 = sval[w][threadIdx.x]; int i2 = sv[w][threadIdx.x];
      if (v2 > bv || (v2 == bv && i2 < bi)) { bv = v2; bi = i2; }
    }
    int b = perm[row0 + threadIdx.x];
    if (b >= 0) bestidx[b] = bi;
  }
}

__global__ void k_out(const int* __restrict__ idx, const int* __restrict__ bestidx,
                      const float* __restrict__ latent, float* __restrict__ out) {
  int b = blockIdx.x;
  int d = threadIdx.x;
  int t = idx[b] & (T_ - 1);
  int v = bestidx[b];
  out[(size_t)b * D_ + d] = latent[(size_t)(t * V_ + v) * D_ + d];
}

extern "C" void kernel_launch(void* const* d_in, const int* in_sizes, int n_in,
                              void* d_out, int out_size, void* d_ws, size_t ws_size,
                              hipStream_t stream) {
  const float* inputs = (const float*)d_in[0];      // (B, D, 1) -> B*D f32
  const float* latent = (const float*)d_in[1];      // (T, V, D) f32
  const int*   index  = (const int*)d_in[2];        // (B,) int

  char* ws = (char*)d_ws;
  int*   counts  = (int*)(ws + WS_COUNTS);
  int*   cursor  = (int*)(ws + WS_CURSOR);
  int*   base    = (int*)(ws + WS_BASE);
  int*   perm    = (int*)(ws + WS_PERM);
  int*   bestidx = (int*)(ws + WS_BESTIDX);
  float* rnorm   = (float*)(ws + WS_RNORM);
  float* xg      = (float*)(ws + WS_XG);
  float* out     = (float*)d_out;

  k_init<<<(CAP_ * D_ + 255) / 256, 256, 0, stream>>>(perm, counts, cursor, xg);
  k_count<<<(B_ + 255) / 256, 256, 0, stream>>>(index, counts);
  k_base<<<1, 1, 0, stream>>>(counts, base);
  k_scatter<<<(B_ + 255) / 256, 256, 0, stream>>>(index, base, cursor, perm);
  k_gatherx<<<CAP_, D_, 0, stream>>>(perm, inputs, xg);
  k_rnorm<<<(T_ * V_) / 8, 256, 0, stream>>>(latent, rnorm);
  k_sim_argmax<<<MAXTILES_, 256, 0, stream>>>(latent, xg, rnorm, base, perm, bestidx);
  k_out<<<B_, D_, 0, stream>>>(index, bestidx, latent, out);
}